// AscendRejectionSampler_11501922419477
// MI455X (gfx1250) — compile-verified
//
#include <hip/hip_runtime.h>
#include <hip/hip_bf16.h>
#include <stdint.h>

// Problem constants (fixed shapes from the reference)
#define BREQ      128        // batch of requests
#define SPEC      4          // max drafts per request
#define OUTW      (SPEC + 1) // output row width
#define VOCAB     128000
#define NTOK      (BREQ * SPEC)
#define PLACEHOLD (-1)

// Argmax streaming config: 25 chunks * 5120 floats = 128000 exactly,
// 5120 = 5 steps * (256 lanes * 4 floats)
#define CHUNKS    25
#define SPAN      5120
#define TPB       256
#define STEPF     1024       // floats per block-step (TPB * 4)
#define NSTEPS    5

// -------- Workspace layout (d_ws) --------
// [0)        : u64 slots[BREQ]   packed argmax keys (atomicMax targets)
// [1024)     : int rejected[BREQ]
// [1536)     : int write_col[BREQ]
// [2048)     : int rec_row[BREQ]

__device__ __forceinline__ unsigned long long shfl_down_u64(unsigned long long x, int off) {
    unsigned lo = (unsigned)x;
    unsigned hi = (unsigned)(x >> 32);
    lo = __shfl_down(lo, off, 32);
    hi = __shfl_down(hi, off, 32);
    return ((unsigned long long)hi << 32) | (unsigned long long)lo;
}

// ---------------- Kernel 1: accept/reject scan + row init ----------------
__global__ __launch_bounds__(BREQ)
void rs_phase1(const float* __restrict__ draft_probs,
               const float* __restrict__ target_probs,
               const float* __restrict__ uniform_probs,
               const int*   __restrict__ draft_token_ids,
               const int*   __restrict__ cu_num_draft,
               const int*   __restrict__ bonus_ids,
               int*          __restrict__ out,
               unsigned long long* __restrict__ slots,
               int* __restrict__ rej, int* __restrict__ wcol, int* __restrict__ rrow) {
    int b = blockIdx.x * blockDim.x + threadIdx.x;
    if (b >= BREQ) return;

    slots[b] = 0ULL;  // re-init atomic slot every call (graph-safe determinism)

    int start = (b == 0) ? 0 : cu_num_draft[b - 1];
    int end   = cu_num_draft[b];
    int n     = end - start;
    if (n > SPEC) n = SPEC;

    float pi = 1.0f, U = 1.0f;
    int last = -1;
    int toks[SPEC];
#pragma unroll
    for (int t = 0; t < SPEC; ++t) toks[t] = PLACEHOLD;

    for (int t = 0; t < n; ++t) {
        int idx = start + t;
        int d   = draft_token_ids[idx];
        size_t base = (size_t)idx * (size_t)VOCAB + (size_t)d;
        float tp = target_probs[base];
        float dp = draft_probs[base];
        float u  = uniform_probs[idx];
        bool  dpos  = (dp > 0.0f);
        float ratio = dpos ? (tp / dp) : 1.0f;
        pi = fminf(pi * ratio, 1.0f);
        U  = U * u;
        if (dpos && (pi >= U)) last = t;
        toks[t] = d;
    }

    bool rejected = (n > 0) && (last != n - 1);

    int* row = out + b * OUTW;
#pragma unroll
    for (int t = 0; t < SPEC; ++t) row[t] = (t <= last) ? toks[t] : PLACEHOLD;
    row[SPEC] = PLACEHOLD;

    int wc = rejected ? (last + 1) : n;
    if (!rejected) row[wc] = bonus_ids[b];

    rej[b]  = rejected ? 1 : 0;
    wcol[b] = wc;
    int rr = start + last + 1;
    rr = rr < 0 ? 0 : (rr > NTOK - 1 ? NTOK - 1 : rr);
    rrow[b] = rr;
}

// ------- Kernel 2: partial argmax over target_probs[rec_row], async LDS stream -------
__global__ __launch_bounds__(TPB)
void rs_argmax(const float* __restrict__ target_probs,
               const int*   __restrict__ rej,
               const int*   __restrict__ rrow,
               unsigned long long* __restrict__ slots) {
    int b     = blockIdx.x / CHUNKS;
    int chunk = blockIdx.x - b * CHUNKS;
    if (rej[b] == 0) return;   // block-uniform skip: no HBM traffic for accepted rows

    const float* rowp = target_probs + (size_t)rrow[b] * (size_t)VOCAB + (size_t)(chunk * SPAN);

    __shared__ float stage[2][STEPF];              // 8 KB double buffer
    __shared__ unsigned long long wbest[TPB / 32];

    const unsigned lane4 = threadIdx.x * 4u;
    // LDS byte offsets for this lane's 16B slot in each buffer.
    // (addrspace(3) offset lives in the low 32 bits of the generic pointer)
    unsigned lds0 = (unsigned)(uintptr_t)&stage[0][lane4];
    unsigned lds1 = (unsigned)(uintptr_t)&stage[1][lane4];

    unsigned long long best = 0ULL;

    // prologue: async-stage step 0
    {
        unsigned long long ga = (unsigned long long)(uintptr_t)(rowp + lane4);
        asm volatile("global_load_async_to_lds_b128 %0, %1, off"
                     :: "v"(lds0), "v"(ga) : "memory");
    }

    for (int s = 0; s < NSTEPS; ++s) {
        if (s + 1 < NSTEPS) {
            unsigned dst = ((s + 1) & 1) ? lds1 : lds0;
            unsigned long long ga =
                (unsigned long long)(uintptr_t)(rowp + (s + 1) * STEPF + lane4);
            asm volatile("global_load_async_to_lds_b128 %0, %1, off"
                         :: "v"(dst), "v"(ga) : "memory");
            asm volatile("s_wait_asynccnt 0x1" ::: "memory"); // step s landed
        } else {
            asm volatile("s_wait_asynccnt 0x0" ::: "memory");
        }

        const float* buf = &stage[s & 1][lane4];   // each lane reads only its own 16B
        unsigned colbase = (unsigned)(chunk * SPAN + s * STEPF) + lane4;
#pragma unroll
        for (int j = 0; j < 4; ++j) {
            float p = buf[j];
            unsigned long long key =
                ((unsigned long long)__float_as_uint(p) << 32) |
                (unsigned long long)(0xFFFFFFFFu - (colbase + (unsigned)j));
            if (key > best) best = key;
        }
    }

    // wave32 reduction
#pragma unroll
    for (int off = 16; off > 0; off >>= 1) {
        unsigned long long o = shfl_down_u64(best, off);
        if (o > best) best = o;
    }
    if ((threadIdx.x & 31u) == 0u) wbest[threadIdx.x >> 5] = best;
    __syncthreads();
    if (threadIdx.x == 0) {
        unsigned long long m = wbest[0];
#pragma unroll
        for (int w = 1; w < TPB / 32; ++w)
            if (wbest[w] > m) m = wbest[w];
        atomicMax(slots + b, m);
    }
}

// ---------------- Kernel 3: decode recovered token ----------------
__global__ __launch_bounds__(BREQ)
void rs_finalize(const unsigned long long* __restrict__ slots,
                 const int* __restrict__ rej,
                 const int* __restrict__ wcol,
                 int* __restrict__ out) {
    int b = blockIdx.x * blockDim.x + threadIdx.x;
    if (b >= BREQ) return;
    if (rej[b]) {
        unsigned long long k = slots[b];
        int rec_id = (int)(0xFFFFFFFFu - (unsigned)(k & 0xFFFFFFFFull));
        out[b * OUTW + wcol[b]] = rec_id;
    }
}

extern "C" void kernel_launch(void* const* d_in, const int* in_sizes, int n_in,
                              void* d_out, int out_size, void* d_ws, size_t ws_size,
                              hipStream_t stream) {
    const float* draft_probs     = (const float*)d_in[0];
    const float* target_probs    = (const float*)d_in[1];
    const float* uniform_probs   = (const float*)d_in[2];
    const int*   draft_token_ids = (const int*)d_in[3];
    const int*   cu_num_draft    = (const int*)d_in[4];
    const int*   bonus_ids       = (const int*)d_in[5];
    int*         out             = (int*)d_out;

    char* ws = (char*)d_ws;
    unsigned long long* slots = (unsigned long long*)(ws);
    int* rej  = (int*)(ws + 1024);
    int* wcol = (int*)(ws + 1536);
    int* rrow = (int*)(ws + 2048);

    rs_phase1<<<1, BREQ, 0, stream>>>(draft_probs, target_probs, uniform_probs,
                                      draft_token_ids, cu_num_draft, bonus_ids,
                                      out, slots, rej, wcol, rrow);

    rs_argmax<<<BREQ * CHUNKS, TPB, 0, stream>>>(target_probs, rej, rrow, slots);

    rs_finalize<<<1, BREQ, 0, stream>>>(slots, rej, wcol, out);
}